// ComputeAlignmentError_927712936255
// MI455X (gfx1250) — compile-verified
//
#include <hip/hip_runtime.h>
#include <hip/hip_bf16.h>
#include <math.h>

typedef __attribute__((ext_vector_type(2))) float v2f;
typedef __attribute__((ext_vector_type(8))) float v8f;

#define NRES   3072
#define NBATCH 2
#define NTILE  (NRES / 16)      // 192 tiles per dimension
#define EPSF   1e-8f
#define WAVES_PER_BLOCK 8
#define IT_PER_WAVE     8       // i-tiles handled per wave (amortize B frags)

// ---------------------------------------------------------------------------
// Frame basis prep: per (b, j) build e1/e2/e3 + fold origin dot-products and
// the reference's +EPS into the K=3 row of the WMMA B operand. Also convert
// the byte mask to a float (0/1) multiplier to keep the epilogue branch-free.
// ---------------------------------------------------------------------------
__device__ __forceinline__ void l2norm3(float &x, float &y, float &z) {
    float n   = sqrtf(fmaf(x, x, fmaf(y, y, z * z)));
    float inv = 1.0f / fmaxf(n, EPSF);
    x *= inv; y *= inv; z *= inv;
}

__device__ __forceinline__ void frame_basis(const float* __restrict__ F,
                                            float e[3][3], float o[3]) {
    // F row-major [3 (dim)][3 (atom)]; atoms: a=col0, b=col1, c=col2
    float ax = F[0], ay = F[3], az = F[6];
    float bx = F[1], by = F[4], bz = F[7];
    float cx = F[2], cy = F[5], cz = F[8];
    float w1x = ax - bx, w1y = ay - by, w1z = az - bz;
    float w2x = cx - bx, w2y = cy - by, w2z = cz - bz;
    l2norm3(w1x, w1y, w1z);
    l2norm3(w2x, w2y, w2z);
    float e1x = w1x + w2x, e1y = w1y + w2y, e1z = w1z + w2z;
    float e2x = w2x - w1x, e2y = w2y - w1y, e2z = w2z - w1z;
    l2norm3(e1x, e1y, e1z);
    l2norm3(e2x, e2y, e2z);
    float e3x = e1y * e2z - e1z * e2y;
    float e3y = e1z * e2x - e1x * e2z;
    float e3z = e1x * e2y - e1y * e2x;
    e[0][0] = e1x; e[0][1] = e1y; e[0][2] = e1z;
    e[1][0] = e2x; e[1][1] = e2y; e[1][2] = e2z;
    e[2][0] = e3x; e[2][1] = e3y; e[2][2] = e3z;
    o[0] = bx; o[1] = by; o[2] = bz;
}

__global__ void frame_prep(const float* __restrict__ predF,
                           const float* __restrict__ trueF,
                           const unsigned char* __restrict__ mask,
                           float* __restrict__ Bp,   // [NBATCH*NRES][3][4]
                           float* __restrict__ Bt,   // [NBATCH*NRES][3][4]
                           float* __restrict__ mf)   // [NBATCH*NRES] 0/1
{
    int idx = blockIdx.x * blockDim.x + threadIdx.x;   // b*NRES + j
    if (idx >= NBATCH * NRES) return;

    float pe[3][3], po[3], te[3][3], to[3];
    frame_basis(predF + (size_t)idx * 9, pe, po);
    frame_basis(trueF + (size_t)idx * 9, te, to);

    float* bp = Bp + (size_t)idx * 12;
    float* bt = Bt + (size_t)idx * 12;
    #pragma unroll
    for (int e = 0; e < 3; ++e) {
        float offP = fmaf(pe[e][0], po[0], fmaf(pe[e][1], po[1], pe[e][2] * po[2]));
        float offT = fmaf(te[e][0], to[0], fmaf(te[e][1], to[1], te[e][2] * to[2]));
        bp[e*4 + 0] =  pe[e][0];
        bp[e*4 + 1] =  pe[e][1];
        bp[e*4 + 2] =  pe[e][2];
        bp[e*4 + 3] =  offT - offP + EPSF;   // K=3 slot: affine + EPS terms
        bt[e*4 + 0] = -te[e][0];             // negated: second WMMA subtracts
        bt[e*4 + 1] = -te[e][1];
        bt[e*4 + 2] = -te[e][2];
        bt[e*4 + 3] =  0.0f;
    }
    mf[idx] = mask[idx] ? 1.0f : 0.0f;
}

// ---------------------------------------------------------------------------
// Main kernel: one wave computes 16x16 (i,j) error tiles via chained
// V_WMMA_F32_16X16X4_F32:   C = A_pred x Bp[e];  S = A_true x Bt[e] + C
// so the accumulator holds s_e = pred_t - true_t + EPS directly.
// Epilogue per element: 1.5 pk-fma + raw v_sqrt + 1 mul + store.
// ---------------------------------------------------------------------------
__global__ void __launch_bounds__(WAVES_PER_BLOCK * 32)
fape_wmma(const float* __restrict__ pred_coords,
          const float* __restrict__ true_coords,
          const float* __restrict__ mf,
          const float* __restrict__ Bp,
          const float* __restrict__ Bt,
          float* __restrict__ out)
{
    const int lane = threadIdx.x & 31;
    const int wave = threadIdx.x >> 5;
    const int gw   = blockIdx.x * WAVES_PER_BLOCK + wave;

    const int itBlocks = NTILE / IT_PER_WAVE;                // 24
    int t = gw;
    const int itb = t % itBlocks;  t /= itBlocks;
    const int jt  = t % NTILE;     t /= NTILE;
    const int b   = t;
    if (b >= NBATCH) return;                                 // uniform guard

    const int lo = lane & 15;
    const int hi = lane >> 4;                                // 0: K=0,1  1: K=2,3
    const int j  = jt * 16 + lo;

    // B fragments (per-wave invariant): float4 per (j, e)
    const float4* bp4 = (const float4*)(Bp + (size_t)(b * NRES + j) * 12);
    const float4* bt4 = (const float4*)(Bt + (size_t)(b * NRES + j) * 12);
    v2f bPf[3], bTf[3];
    #pragma unroll
    for (int e = 0; e < 3; ++e) {
        float4 vp = bp4[e];
        float4 vt = bt4[e];
        bPf[e].x = hi ? vp.z : vp.x;  bPf[e].y = hi ? vp.w : vp.y;
        bTf[e].x = hi ? vt.z : vt.x;  bTf[e].y = hi ? vt.w : vt.y;
    }
    const float mj = mf[b * NRES + j];

    for (int it = 0; it < IT_PER_WAVE; ++it) {
        const int i0 = (itb * IT_PER_WAVE + it) * 16;
        const int iA = i0 + lo;                              // A-operand row
        const float* pc = pred_coords + (size_t)(b * NRES + iA) * 3;
        const float* tc = true_coords + (size_t)(b * NRES + iA) * 3;

        v2f aP, aT;                                          // 16x4 f32 A frags
        aP.x = pc[2 * hi];  aP.y = hi ? 1.0f : pc[1];        // K=3 row == 1.0
        aT.x = tc[2 * hi];  aT.y = hi ? 0.0f : tc[1];        // K=3 row == 0.0

        v8f s[3];
        #pragma unroll
        for (int e = 0; e < 3; ++e) {
            v8f c = {};
            c = __builtin_amdgcn_wmma_f32_16x16x4_f32(
                    false, aP, false, bPf[e], (short)0, c, false, false);
            s[e] = __builtin_amdgcn_wmma_f32_16x16x4_f32(
                    false, aT, false, bTf[e], (short)0, c, false, false);
        }

        // Epilogue: ||s||_2 per element, float-mask multiply, store
        #pragma unroll
        for (int r = 0; r < 8; ++r) {
            const int i = i0 + r + 8 * hi;                   // C/D row layout
            float q = s[0][r] * s[0][r];
            q = fmaf(s[1][r], s[1][r], q);
            q = fmaf(s[2][r], s[2][r], q);
            float err = __builtin_amdgcn_sqrtf(q);           // raw v_sqrt_f32
            float m   = mf[b * NRES + i] * mj;               // 0/1 pair mask
            out[((size_t)(b * NRES + i)) * NRES + j] = err * m;
        }
    }
}

// ---------------------------------------------------------------------------
extern "C" void kernel_launch(void* const* d_in, const int* in_sizes, int n_in,
                              void* d_out, int out_size, void* d_ws, size_t ws_size,
                              hipStream_t stream) {
    const float* pred_coords = (const float*)d_in[0];
    const float* true_coords = (const float*)d_in[1];
    const float* pred_frames = (const float*)d_in[2];
    const float* true_frames = (const float*)d_in[3];
    const unsigned char* mask = (const unsigned char*)d_in[4];
    float* out = (float*)d_out;

    // Workspace: Bp | Bt | mf  (288KB + 288KB + 24KB)
    float* Bp = (float*)d_ws;
    float* Bt = Bp + (size_t)NBATCH * NRES * 12;
    float* mf = Bt + (size_t)NBATCH * NRES * 12;

    const int prepThreads = NBATCH * NRES;
    frame_prep<<<(prepThreads + 255) / 256, 256, 0, stream>>>(
        pred_frames, true_frames, mask, Bp, Bt, mf);

    const int totalWaves = NBATCH * NTILE * (NTILE / IT_PER_WAVE);   // 9216
    fape_wmma<<<totalWaves / WAVES_PER_BLOCK, WAVES_PER_BLOCK * 32, 0, stream>>>(
        pred_coords, true_coords, mf, Bp, Bt, out);
}